// GaussianMixtureModel_60395830116897
// MI455X (gfx1250) — compile-verified
//
#include <hip/hip_runtime.h>
#include <stdint.h>

#define NCOMP 16
#define NFEAT 32

typedef float v2f __attribute__((ext_vector_type(2)));
typedef float v8f __attribute__((ext_vector_type(8)));

// ---------------- workspace layout ----------------
// bytes [0, 65536)        : L[16][32][32] f32 (row-major, lower-triangular)
// ints  meta = ws + 16384 : [0..15] counts, [16..31] bucket offsets,
//                           [32..47] scatter cursors, [48..64] padded-tile
//                           offsets (17 entries), cdf (floats) at [80..95]
// ints  perm = meta + 128 : N ints, bucket-ordered original sample ids
// total required: 65536 + 512 + 4*N bytes

static __device__ __forceinline__ uint32_t mix32(uint32_t x) {
  x ^= x >> 16; x *= 0x7feb352dU;
  x ^= x >> 15; x *= 0x846ca68bU;
  x ^= x >> 16; return x;
}
static __device__ __forceinline__ uint32_t rng2(uint32_t i, uint32_t j) {
  return mix32(mix32(i * 2654435761u + j * 2246822519u + 0x9E3779B9u) + j);
}
static __device__ __forceinline__ float u01(uint32_t h) {
  return (float)((h >> 8) + 1u) * (1.0f / 16777216.0f);  // (0,1]
}
// categorical draw for sample i against precomputed normalized CDF
static __device__ __forceinline__ int pick_k(uint32_t i, const float* __restrict__ cdf) {
  float u = u01(rng2(i, 1000003u));
  int k = 0;
#pragma unroll
  for (int t = 0; t < NCOMP - 1; ++t) k += (u >= cdf[t]) ? 1 : 0;
  return k;
}
// Box-Muller pair p (columns 2p, 2p+1) of sample i
static __device__ __forceinline__ void bm_pair(uint32_t i, int p, float* z0, float* z1) {
  float u1 = u01(rng2(i, (uint32_t)(2 * p)));
  float u2 = u01(rng2(i, (uint32_t)(2 * p + 1)));
  float r = __fsqrt_rn(-2.0f * __logf(u1));
  float sn, cs;
  __sincosf(6.28318530718f * u2, &sn, &cs);
  *z0 = r * cs;
  *z1 = r * sn;
}

// -------- 1) per-component 32x32 Cholesky (16 blocks x 32 threads) --------
__global__ void cholesky_kernel(const float* __restrict__ covs, float* __restrict__ Lg) {
  __shared__ float a[NFEAT][NFEAT + 1];
  const int k = blockIdx.x, t = threadIdx.x;
  for (int j = 0; j < NFEAT; ++j) a[t][j] = covs[(k * NFEAT + t) * NFEAT + j];
  __syncthreads();
  for (int j = 0; j < NFEAT; ++j) {
    if (t == j) a[j][j] = sqrtf(a[j][j]);
    __syncthreads();
    if (t > j) a[t][j] /= a[j][j];
    __syncthreads();
    if (t > j)
      for (int l = j + 1; l <= t; ++l) a[t][l] -= a[t][j] * a[l][j];
    __syncthreads();
  }
  for (int j = 0; j < NFEAT; ++j)
    Lg[(k * NFEAT + t) * NFEAT + j] = (j <= t) ? a[t][j] : 0.0f;
}

// -------- 2) zero counters + build normalized CDF (every call) --------
__global__ void init_meta_kernel(const float* __restrict__ w, int* __restrict__ meta) {
  const int t = threadIdx.x;
  if (t < NCOMP) meta[t] = 0;
  if (t == 0) {
    float s = 0.f;
    for (int k = 0; k < NCOMP; ++k) s += w[k];
    const float inv = 1.0f / s;
    float acc = 0.f;
    float* cdf = (float*)(meta + 80);
    for (int k = 0; k < NCOMP; ++k) { acc += w[k] * inv; cdf[k] = acc; }
  }
}

// -------- 3) histogram of component assignments (warp-aggregated) --------
__global__ void count_kernel(int N, int* __restrict__ meta) {
  const float* cdf = (const float*)(meta + 80);
  const int gid = blockIdx.x * blockDim.x + threadIdx.x;
  const int stride = gridDim.x * blockDim.x;
  const int Nr = (N + 31) & ~31;
  const int lane = threadIdx.x & 31;
  for (int i = gid; i < Nr; i += stride) {
    const bool valid = i < N;
    const int myk = valid ? pick_k((uint32_t)i, cdf) : -1;
#pragma unroll
    for (int k = 0; k < NCOMP; ++k) {
      unsigned long long m = __ballot(myk == k);
      if (m) {
        const int leader = __ffsll(m) - 1;
        if (lane == leader) atomicAdd(&meta[k], (int)__popcll(m));
      }
    }
  }
}

// -------- 4) tiny scan: bucket offsets, cursors, padded-tile offsets --------
__global__ void scan_kernel(int* __restrict__ meta) {
  if (threadIdx.x == 0 && blockIdx.x == 0) {
    int tot = 0, pt = 0;
    for (int k = 0; k < NCOMP; ++k) {
      meta[16 + k] = tot; tot += meta[k];
      meta[32 + k] = 0;
      meta[48 + k] = pt; pt += (meta[k] + 15) >> 4;
    }
    meta[48 + 16] = pt;  // total padded 16-row tiles
  }
}

// -------- 5) build bucket-ordered permutation (warp-aggregated atomics) ----
__global__ void scatter_kernel(int N, int* __restrict__ meta, int* __restrict__ perm) {
  const float* cdf = (const float*)(meta + 80);
  const int gid = blockIdx.x * blockDim.x + threadIdx.x;
  const int stride = gridDim.x * blockDim.x;
  const int Nr = (N + 31) & ~31;
  const int lane = threadIdx.x & 31;
  const unsigned long long below = (lane == 0) ? 0ull : ((1ull << lane) - 1ull);
  for (int i = gid; i < Nr; i += stride) {
    const bool valid = i < N;
    const int myk = valid ? pick_k((uint32_t)i, cdf) : -1;
#pragma unroll
    for (int k = 0; k < NCOMP; ++k) {
      unsigned long long m = __ballot(myk == k);
      if (m) {
        const int leader = __ffsll(m) - 1;
        int base = 0;
        if (lane == leader) base = atomicAdd(&meta[32 + k], (int)__popcll(m));
        base = __shfl(base, leader);
        if (myk == k) {
          const int rank = (int)__popcll(m & below);
          perm[meta[16 + k] + base + rank] = i;
        }
      }
    }
  }
}

// -------- 6) main kernel: per-bucket 16x32 tiles via V_WMMA_F32_16X16X4_F32 --
__global__ void __launch_bounds__(256) gmm_wmma_kernel(
    const float* __restrict__ Lg, const float* __restrict__ means,
    const int* __restrict__ meta, const int* __restrict__ perm,
    float* __restrict__ out) {
  __shared__ __align__(16) float ldsL[NCOMP * NFEAT * NFEAT];  // 64 KB
  // stage all 16 Cholesky factors into LDS
  for (int i = threadIdx.x; i < NCOMP * NFEAT * NFEAT / 4; i += blockDim.x)
    ((float4*)ldsL)[i] = ((const float4*)Lg)[i];
  __syncthreads();

  const int totalTiles = meta[48 + 16];
  const int w = (blockIdx.x * blockDim.x + threadIdx.x) >> 5;
  const int nw = (gridDim.x * blockDim.x) >> 5;
  const int lane = threadIdx.x & 31;
  const int m = lane & 15;   // A: row M; B/C/D: column N
  const int hi = lane >> 4;  // lane-half selects K pair / M-half

  for (int tile = w; tile < totalTiles; tile += nw) {
    // locate bucket k and local row base r0 for this padded tile
    int k = 0;
#pragma unroll
    for (int t = 1; t < NCOMP; ++t) k = (tile >= meta[48 + t]) ? t : k;
    const int r0 = (tile - meta[48 + k]) << 4;
    const int cnt = meta[k];
    const int offk = meta[16 + k];

    // A-phase: this lane supplies z[sample m][4s+2*hi .. +1] for s=0..7
    const int g = r0 + m;
    const bool valid = g < cnt;
    const uint32_t orig = valid ? (uint32_t)perm[offk + g] : 0u;
    float az0[8], az1[8];
#pragma unroll
    for (int s = 0; s < 8; ++s) {
      const int p = 2 * s + hi;  // columns 2p, 2p+1 == 4s+2hi, 4s+2hi+1
      float z0, z1;
      bm_pair(orig, p, &z0, &z1);
      az0[s] = valid ? z0 : 0.0f;
      az1[s] = valid ? z1 : 0.0f;
    }

    v8f c0 = {0.f, 0.f, 0.f, 0.f, 0.f, 0.f, 0.f, 0.f};  // features 0..15
    v8f c1 = {0.f, 0.f, 0.f, 0.f, 0.f, 0.f, 0.f, 0.f};  // features 16..31
#pragma unroll
    for (int s = 0; s < 8; ++s) {
      v2f a; a[0] = az0[s]; a[1] = az1[s];
      const int j0 = 4 * s + 2 * hi;
      // B[kk][n] = L[k][fbase+n][4s+kk]  (z @ L^T); lane column n = m
      const float* bp0 = &ldsL[(k * NFEAT + m) * NFEAT + j0];
      const float* bp1 = &ldsL[(k * NFEAT + 16 + m) * NFEAT + j0];
      v2f b0; b0[0] = bp0[0]; b0[1] = bp0[1];
      v2f b1; b1[0] = bp1[0]; b1[1] = bp1[1];
      c0 = __builtin_amdgcn_wmma_f32_16x16x4_f32(false, a, false, b0,
                                                 (short)0, c0, false, false);
      c1 = __builtin_amdgcn_wmma_f32_16x16x4_f32(false, a, false, b1,
                                                 (short)0, c1, false, false);
    }

    // epilogue: D vgpr i holds row M = 8*hi + i, column N = m
    const int f0 = m, f1 = 16 + m;
    const float mv0 = means[k * NFEAT + f0];
    const float mv1 = means[k * NFEAT + f1];
#pragma unroll
    for (int i2 = 0; i2 < 8; ++i2) {
      const int g2 = r0 + hi * 8 + i2;
      if (g2 < cnt) {
        const int o2 = perm[offk + g2];
        out[o2 * NFEAT + f0] = mv0 + c0[i2];
        out[o2 * NFEAT + f1] = mv1 + c1[i2];
      }
    }
  }
}

// -------- fallback (ws too small for permutation): plain VALU path --------
__global__ void __launch_bounds__(256) gmm_direct_kernel(
    int N, const float* __restrict__ Lg, const float* __restrict__ means,
    const int* __restrict__ meta, float* __restrict__ out) {
  __shared__ __align__(16) float ldsL[NCOMP * NFEAT * NFEAT];
  for (int i = threadIdx.x; i < NCOMP * NFEAT * NFEAT / 4; i += blockDim.x)
    ((float4*)ldsL)[i] = ((const float4*)Lg)[i];
  __syncthreads();
  const float* cdf = (const float*)(meta + 80);
  const int gid = blockIdx.x * blockDim.x + threadIdx.x;
  const int stride = gridDim.x * blockDim.x;
  for (int i = gid; i < N; i += stride) {
    const int k = pick_k((uint32_t)i, cdf);
    float z[NFEAT];
#pragma unroll
    for (int p = 0; p < NFEAT / 2; ++p) bm_pair((uint32_t)i, p, &z[2 * p], &z[2 * p + 1]);
#pragma unroll
    for (int f = 0; f < NFEAT; ++f) {
      float acc = means[k * NFEAT + f];
      for (int j = 0; j <= f; ++j) acc += ldsL[(k * NFEAT + f) * NFEAT + j] * z[j];
      out[i * NFEAT + f] = acc;
    }
  }
}

extern "C" void kernel_launch(void* const* d_in, const int* in_sizes, int n_in,
                              void* d_out, int out_size, void* d_ws, size_t ws_size,
                              hipStream_t stream) {
  const float* d_w     = (const float*)d_in[1];
  const float* d_means = (const float*)d_in[2];
  const float* d_covs  = (const float*)d_in[3];
  float* out = (float*)d_out;

  float* Lg  = (float*)d_ws;
  int*  meta = (int*)d_ws + 16384;   // after 64 KB of L
  int*  perm = meta + 128;           // after 512 B of metadata

  const int N = out_size / NFEAT;    // out is [N, 32] f32
  const size_t ws_need = 65536u + 512u + (size_t)N * 4u;

  int blocksCS = (N + 255) / 256;
  if (blocksCS > 4096) blocksCS = 4096;
  if (blocksCS < 1) blocksCS = 1;

  cholesky_kernel<<<NCOMP, NFEAT, 0, stream>>>(d_covs, Lg);
  init_meta_kernel<<<1, 32, 0, stream>>>(d_w, meta);

  if (ws_size >= ws_need) {
    count_kernel<<<blocksCS, 256, 0, stream>>>(N, meta);
    scan_kernel<<<1, 1, 0, stream>>>(meta);
    scatter_kernel<<<blocksCS, 256, 0, stream>>>(N, meta, perm);
    int blocksG = 2048;
    gmm_wmma_kernel<<<blocksG, 256, 0, stream>>>(Lg, d_means, meta, perm, out);
  } else {
    gmm_direct_kernel<<<blocksCS, 256, 0, stream>>>(N, Lg, d_means, meta, out);
  }
}